// S5_19842748908223
// MI455X (gfx1250) — compile-verified
//
#include <hip/hip_runtime.h>
#include <hip/hip_bf16.h>

typedef float v2f __attribute__((ext_vector_type(2)));
typedef float v8f __attribute__((ext_vector_type(8)));

typedef __attribute__((address_space(1))) void g_void;
typedef __attribute__((address_space(3))) void l_void;

// async-to-LDS builtin takes pointers to 16-byte int vectors (per diagnostic)
typedef int b128_t __attribute__((vector_size(16)));
typedef __attribute__((address_space(1))) b128_t* gp128;
typedef __attribute__((address_space(3))) b128_t* lp128;

#define HALF   32
#define DM     2048
#define BB     4
#define LL     4096
#define MM     (BB*LL)        // 16384 rows (b*l)
#define CHUNK  128
#define NCHUNK (LL/CHUNK)     // 32 chunks per sequence

// ---- workspace layout (float offsets) ----
#define OFF_LAMR   0                         // lam_bar re [32]
#define OFF_LAMI   32                        // lam_bar im [32]
#define OFF_BFR    64                        // (lam_bar-1)/lam re [32]
#define OFF_BFI    96                        // im [32]
#define OFF_PWRE   128                       // lam_bar^t re [32][129]
#define OFF_PWIM   (OFF_PWRE + 32*129)       // 4256
#define OFF_BBAR   (OFF_PWIM + 32*129)       // 8384 : B_bar [64][2048] rows 0..31 re, 32..63 im
#define OFF_W      (OFF_BBAR + 64*DM)        // 139456 : W [64][2048] folded output weights
#define OFF_BU     (OFF_W + 64*DM)           // 270528 : Bu / xs [M][64] (cols 0..31 re, 32..63 im)
#define OFF_CARA   (OFF_BU + (size_t)MM*64)  // chunk-local end state [b][c][n], re then im
#define OFF_CARP   (OFF_CARA + 2*BB*NCHUNK*HALF)
#define CARSZ      (BB*NCHUNK*HALF)          // 4096

// ---- CDNA5 async global->LDS copy (ASYNCcnt-tracked), with asm fallback ----
__device__ __forceinline__ void async_copy_b128(const float* gptr, float* lptr) {
#if __has_builtin(__builtin_amdgcn_global_load_async_to_lds_b128)
    __builtin_amdgcn_global_load_async_to_lds_b128(
        (gp128)(g_void*)gptr, (lp128)(l_void*)lptr, 0, 0);
#else
    unsigned laddr = (unsigned)(uintptr_t)(l_void*)lptr;
    asm volatile("global_load_async_to_lds_b128 %0, %1, off"
                 :: "v"(laddr), "v"(gptr) : "memory");
#endif
}

#if __has_builtin(__builtin_amdgcn_s_wait_asynccnt)
#define WAIT_ASYNC(n) __builtin_amdgcn_s_wait_asynccnt(n)
#else
#define WAIT_ASYNC(n) asm volatile("s_wait_asynccnt %0" :: "i"(n) : "memory")
#endif

// -------------------------------------------------------------------
// Kernel 1: per-state setup. softplus -> Lambda, ZOH discretization,
// B_bar factor, and table of lam_bar^t for t=0..128 (for chunked scan).
// -------------------------------------------------------------------
__global__ void k_setup(const float* __restrict__ Lur, const float* __restrict__ Lim,
                        const float* __restrict__ logD, float* __restrict__ ws) {
    int n = threadIdx.x;
    if (n >= HALF) return;
    float x   = Lur[n];
    float sp  = (x > 20.f) ? x : log1pf(expf(x));
    float lre = -(sp + 1e-4f + 0.01f);
    float lim = Lim[n];
    float dt  = expf(logD[n]);
    float e   = expf(lre * dt);
    float lbr = e * cosf(lim * dt);
    float lbi = e * sinf(lim * dt);
    ws[OFF_LAMR + n] = lbr;
    ws[OFF_LAMI + n] = lbi;
    float den = lre * lre + lim * lim;
    float nr = lbr - 1.f, ni = lbi;
    ws[OFF_BFR + n] = (nr * lre + ni * lim) / den;
    ws[OFF_BFI + n] = (ni * lre - nr * lim) / den;
    float pr = 1.f, pi = 0.f;
    ws[OFF_PWRE + n * 129] = 1.f;
    ws[OFF_PWIM + n * 129] = 0.f;
    for (int t = 1; t <= 128; ++t) {
        float prn = pr * lbr - pi * lbi;
        float pin = pr * lbi + pi * lbr;
        pr = prn; pi = pin;
        ws[OFF_PWRE + n * 129 + t] = pr;
        ws[OFF_PWIM + n * 129 + t] = pi;
    }
}

// -------------------------------------------------------------------
// Kernel 2: B_bar[n,d] = bfac[n] * (B_re + i B_im)[n,d]  -> [64][2048]
// -------------------------------------------------------------------
__global__ void k_bbar(const float* __restrict__ Bre, const float* __restrict__ Bim,
                       float* __restrict__ ws) {
    int idx = blockIdx.x * blockDim.x + threadIdx.x;
    if (idx >= HALF * DM) return;
    int n = idx / DM, d = idx % DM;
    float fr = ws[OFF_BFR + n], fi = ws[OFF_BFI + n];
    float br = Bre[idx], bi = Bim[idx];
    ws[OFF_BBAR + (size_t)n * DM + d]          = fr * br - fi * bi;   // re
    ws[OFF_BBAR + (size_t)(HALF + n) * DM + d] = fr * bi + fi * br;   // im
}

// -------------------------------------------------------------------
// Kernel 3: W[j,d] : j<32 -> 2*C_re[d,j] ; j>=32 -> -2*C_im[d,j-32]
// -------------------------------------------------------------------
__global__ void k_w(const float* __restrict__ Cre, const float* __restrict__ Cim,
                    float* __restrict__ ws) {
    int idx = blockIdx.x * blockDim.x + threadIdx.x;
    if (idx >= 64 * DM) return;
    int j = idx / DM, d = idx % DM;
    float v = (j < HALF) ? 2.f * Cre[(size_t)d * HALF + j]
                         : -2.f * Cim[(size_t)d * HALF + (j - HALF)];
    ws[OFF_W + (size_t)j * DM + d] = v;
}

// -------------------------------------------------------------------
// Kernel 4: GEMM1  Bu[M][64] = u[M][2048] @ B_bar^T, f32 WMMA 16x16x4.
// Double-buffered async global->LDS staging (ASYNCcnt), K-slice = 64.
// Block = 256 thr = 8 waves (2 M-tiles x 4 N-tiles), M-tile 32 rows.
// -------------------------------------------------------------------
#define KS     64
#define NSLICE (DM/KS)   // 32
#define OPS    6         // async b128 issues per thread per slice (2 u + 4 Bbar)

__global__ __launch_bounds__(256) void k_gemm1(const float* __restrict__ u,
                                               float* __restrict__ ws) {
    __shared__ float As[2][32][68];  // u tile    [buf][32 rows][64 k] (+pad)
    __shared__ float Bs[2][64][68];  // Bbar tile [buf][64 rows][64 k] (+pad)
    const float* Bbar = ws + OFF_BBAR;
    float* bu = ws + OFF_BU;
    int tid  = threadIdx.x;
    int wave = tid >> 5, lane = tid & 31;
    int m0   = blockIdx.x * 32;
    int moff = (wave >> 2) * 16;   // 0 or 16
    int joff = (wave & 3) * 16;    // 0,16,32,48
    int row  = lane & 15;
    int koff = (lane >> 4) * 2;    // lanes 16..31 carry K=2,3
    v8f acc = {0.f, 0.f, 0.f, 0.f, 0.f, 0.f, 0.f, 0.f};

    auto issue = [&](int s, int buf) {
        int ks = s * KS;
#pragma unroll
        for (int i = 0; i < 2; ++i) {            // u tile: 512 float4
            int lin = tid + i * 256;
            int r = lin >> 4, c4 = lin & 15;
            async_copy_b128(u + (size_t)(m0 + r) * DM + ks + c4 * 4, &As[buf][r][c4 * 4]);
        }
#pragma unroll
        for (int i = 0; i < 4; ++i) {            // Bbar tile: 1024 float4
            int lin = tid + i * 256;
            int r = lin >> 4, c4 = lin & 15;
            async_copy_b128(Bbar + (size_t)r * DM + ks + c4 * 4, &Bs[buf][r][c4 * 4]);
        }
    };

    issue(0, 0);
    for (int s = 0; s < NSLICE; ++s) {
        int buf = s & 1;
        if (s + 1 < NSLICE) {
            issue(s + 1, buf ^ 1);   // prefetch next slice into other buffer
            WAIT_ASYNC(OPS);         // retire slice s (async ops complete in order)
        } else {
            WAIT_ASYNC(0);
        }
        __syncthreads();             // slice-s LDS data visible to all waves
#pragma unroll
        for (int kk = 0; kk < KS; kk += 4) {
            v2f a = *(const v2f*)&As[buf][moff + row][kk + koff];
            v2f b = *(const v2f*)&Bs[buf][joff + row][kk + koff];
            acc = __builtin_amdgcn_wmma_f32_16x16x4_f32(false, a, false, b,
                                                        (short)0, acc, false, false);
        }
        __syncthreads();             // done reading buf before it is refilled
    }
    int mbase = m0 + moff + ((lane >> 4) * 8);
    for (int r = 0; r < 8; ++r)
        bu[(size_t)(mbase + r) * 64 + joff + row] = acc[r];
}

// -------------------------------------------------------------------
// Kernel 5: chunked scan pass A — local scan of each 128-step chunk.
// -------------------------------------------------------------------
__global__ void k_scanA(float* __restrict__ ws) {
    int gid = blockIdx.x * blockDim.x + threadIdx.x;   // 0..4095
    int b = gid >> 10;
    int c = (gid >> 5) & 31;
    int n = gid & 31;
    float lbr = ws[OFF_LAMR + n], lbi = ws[OFF_LAMI + n];
    float* bu = ws + OFF_BU;
    size_t m0 = (size_t)b * LL + (size_t)c * CHUNK;
    float xr = 0.f, xi = 0.f;
    for (int t = 0; t < CHUNK; ++t) {
        float* p = bu + (m0 + t) * 64;
        float br = p[n], bi = p[32 + n];
        float nxr = lbr * xr - lbi * xi + br;
        float nxi = lbr * xi + lbi * xr + bi;
        xr = nxr; xi = nxi;
        p[n] = xr; p[32 + n] = xi;
    }
    int ci = (b * NCHUNK + c) * HALF + n;
    ws[OFF_CARA + ci]         = xr;
    ws[OFF_CARA + CARSZ + ci] = xi;
}

// -------------------------------------------------------------------
// Kernel 6: scan pass B — serial scan of 32 chunk carries per (b,n).
// -------------------------------------------------------------------
__global__ void k_scanB(float* __restrict__ ws) {
    int gid = threadIdx.x;            // blockDim = 128
    if (gid >= BB * HALF) return;
    int b = gid >> 5, n = gid & 31;
    float lr = ws[OFF_PWRE + n * 129 + 128];
    float li = ws[OFF_PWIM + n * 129 + 128];
    float cr = 0.f, ci = 0.f;
    for (int c = 0; c < NCHUNK; ++c) {
        int idx = (b * NCHUNK + c) * HALF + n;
        float ar = ws[OFF_CARA + idx], ai = ws[OFF_CARA + CARSZ + idx];
        float ncr = lr * cr - li * ci + ar;
        float nci = lr * ci + li * cr + ai;
        cr = ncr; ci = nci;
        ws[OFF_CARP + idx]         = cr;
        ws[OFF_CARP + CARSZ + idx] = ci;
    }
}

// -------------------------------------------------------------------
// Kernel 7: fix-up — x_global = x_local + lam_bar^{t+1} * carry_{c-1}
// -------------------------------------------------------------------
__global__ void k_fix(float* __restrict__ ws) {
    int gid = blockIdx.x * blockDim.x + threadIdx.x;   // M*32 threads
    if (gid >= MM * HALF) return;
    int m = gid >> 5, n = gid & 31;
    int l = m & (LL - 1);
    int c = l >> 7;
    if (c == 0) return;
    int b = m >> 12;
    int t = l & 127;
    float pr = ws[OFF_PWRE + n * 129 + t + 1];
    float pi = ws[OFF_PWIM + n * 129 + t + 1];
    int ci = (b * NCHUNK + (c - 1)) * HALF + n;
    float cr = ws[OFF_CARP + ci], cm = ws[OFF_CARP + CARSZ + ci];
    float* p = ws + OFF_BU + (size_t)m * 64;
    float xr = p[n], xi = p[32 + n];
    p[n]      = xr + pr * cr - pi * cm;
    p[32 + n] = xi + pr * cm + pi * cr;
}

// -------------------------------------------------------------------
// Kernel 8: GEMM2  y[M][2048] = xs[M][64] @ W[64][2048] + D*u
// -------------------------------------------------------------------
__global__ __launch_bounds__(256) void k_gemm2(const float* __restrict__ u,
                                               const float* __restrict__ Dv,
                                               float* __restrict__ out,
                                               const float* __restrict__ ws) {
    __shared__ float Xs[16][68];
    const float* W  = ws + OFF_W;
    const float* bu = ws + OFF_BU;
    int tid = threadIdx.x;
    int m0  = blockIdx.x * 16;
    int nb  = blockIdx.y * 512;
    {
        int r = tid >> 4, c4 = tid & 15;
        float4 v = *(const float4*)(bu + (size_t)(m0 + r) * 64 + c4 * 4);
        Xs[r][c4 * 4 + 0] = v.x; Xs[r][c4 * 4 + 1] = v.y;
        Xs[r][c4 * 4 + 2] = v.z; Xs[r][c4 * 4 + 3] = v.w;
    }
    __syncthreads();
    int wave = tid >> 5, lane = tid & 31;
    int row  = lane & 15;
    int koff = (lane >> 4) * 2;
    int nw   = nb + wave * 64;
    for (int t = 0; t < 4; ++t) {
        int nt = nw + t * 16;
        v8f acc = {0.f, 0.f, 0.f, 0.f, 0.f, 0.f, 0.f, 0.f};
#pragma unroll
        for (int kk = 0; kk < 64; kk += 4) {
            int k0 = kk + koff;
            v2f a = *(const v2f*)&Xs[row][k0];
            v2f b;
            b.x = W[(size_t)k0 * DM + nt + row];
            b.y = W[(size_t)(k0 + 1) * DM + nt + row];
            acc = __builtin_amdgcn_wmma_f32_16x16x4_f32(false, a, false, b,
                                                        (short)0, acc, false, false);
        }
        int hi = (lane >> 4) * 8;
        for (int r = 0; r < 8; ++r) {
            size_t m = (size_t)(m0 + hi + r);
            size_t d = (size_t)(nt + row);
            out[m * DM + d] = acc[r] + Dv[d] * u[m * DM + d];
        }
    }
}

// -------------------------------------------------------------------
extern "C" void kernel_launch(void* const* d_in, const int* in_sizes, int n_in,
                              void* d_out, int out_size, void* d_ws, size_t ws_size,
                              hipStream_t stream) {
    (void)in_sizes; (void)n_in; (void)out_size; (void)ws_size;
    const float* u    = (const float*)d_in[0];
    const float* Lur  = (const float*)d_in[1];
    const float* Lim  = (const float*)d_in[2];
    const float* Bre  = (const float*)d_in[3];
    const float* Bim  = (const float*)d_in[4];
    const float* Cre  = (const float*)d_in[5];
    const float* Cim  = (const float*)d_in[6];
    const float* Dv   = (const float*)d_in[7];
    const float* logD = (const float*)d_in[8];
    float* out = (float*)d_out;
    float* ws  = (float*)d_ws;

    k_setup<<<1, 32, 0, stream>>>(Lur, Lim, logD, ws);
    k_bbar <<<(HALF * DM + 255) / 256, 256, 0, stream>>>(Bre, Bim, ws);
    k_w    <<<(64 * DM + 255) / 256, 256, 0, stream>>>(Cre, Cim, ws);
    k_gemm1<<<MM / 32, 256, 0, stream>>>(u, ws);
    k_scanA<<<16, 256, 0, stream>>>(ws);
    k_scanB<<<1, 128, 0, stream>>>(ws);
    k_fix  <<<(MM * HALF) / 256, 256, 0, stream>>>(ws);
    k_gemm2<<<dim3(MM / 16, 4), 256, 0, stream>>>(u, Dv, out, ws);
}